// SSM_5643587027233
// MI455X (gfx1250) — compile-verified
//
#include <hip/hip_runtime.h>
#include <math.h>

#define BSZ 128
#define DD  12
#define NN  12
#define LL  4037
#define TL  32          // L-tile; two 16-row WMMA M-blocks
#define NCH 8           // L-chunks for the 3-phase parallel scan
#define CHL 505         // ceil(LL / NCH)
#define DN  (DD * NN)   // 144 recurrences per (b)

typedef __attribute__((ext_vector_type(2))) float v2f;
typedef __attribute__((ext_vector_type(8))) float v8f;

// ---- CDNA5 async global->LDS staging (ASYNCcnt) with synchronous fallback ----
#if defined(__HIP_DEVICE_COMPILE__) && \
    __has_builtin(__builtin_amdgcn_global_load_async_to_lds_b32) && \
    __has_builtin(__builtin_amdgcn_s_wait_asynccnt)
#define USE_ASYNC 1
#else
#define USE_ASYNC 0
#endif

typedef float XRow[DD + 1];

__device__ __forceinline__ void stage_tile(const float* __restrict__ xb,
                                           XRow* xs, int l0, int tid) {
  for (int idx = tid; idx < DD * TL; idx += 256) {
    const int d = idx >> 5;            // idx / TL
    const int i = idx & (TL - 1);
    const int l = l0 + i;
#if USE_ASYNC
    if (l < LL) {
      __builtin_amdgcn_global_load_async_to_lds_b32(
          (__attribute__((address_space(1))) int*)(xb + (size_t)d * LL + l),
          (__attribute__((address_space(3))) int*)&xs[i][d],
          /*offset=*/0, /*cpol=*/0);
    } else {
      xs[i][d] = 0.f;
    }
#else
    xs[i][d] = (l < LL) ? xb[(size_t)d * LL + l] : 0.f;
    const int lp = l + TL;
    if (lp < LL) __builtin_prefetch(&xb[(size_t)d * LL + lp], 0, 1);
#endif
  }
}

__device__ __forceinline__ void wait_stage() {
#if USE_ASYNC
  __builtin_amdgcn_s_wait_asynccnt(0);
#endif
}

__device__ __forceinline__ v8f wmma4(v2f a, v2f b, v8f c) {
  return __builtin_amdgcn_wmma_f32_16x16x4_f32(false, a, false, b, (short)0, c,
                                               false, false);
}

// =======================  K1: local scan per (b, chunk)  =======================
__global__ __launch_bounds__(256)
void ssm_local_kernel(const float* __restrict__ x,
                      const float* __restrict__ A,
                      const float* __restrict__ Bw, const float* __restrict__ Bb,
                      const float* __restrict__ Cw, const float* __restrict__ Cb,
                      const float* __restrict__ dw, const float* __restrict__ dbp,
                      float* __restrict__ out,
                      float* __restrict__ hfin, float* __restrict__ Pfin)
{
  __shared__ float xs[2][TL][DD + 1];
  __shared__ float BmS[TL][16];
  __shared__ float CmS[TL][16];
  __shared__ float deltaS[TL];
  __shared__ float hist[TL][DN];
  __shared__ float wBw[NN][DD], wCw[NN][DD];
  __shared__ float wA[DN];
  __shared__ float wBb[NN], wCb[NN], wdw[DD];
  __shared__ float wdb;

  const int cc  = blockIdx.x;                    // chunk
  const int b   = blockIdx.y;                    // batch
  const int tid = threadIdx.x;
  const int lbeg = cc * CHL;
  const int lend = (lbeg + CHL < LL) ? lbeg + CHL : LL;

  if (tid < DN) {
    wBw[tid / DD][tid % DD] = Bw[tid];
    wCw[tid / DD][tid % DD] = Cw[tid];
    wA[tid] = A[tid];
  }
  if (tid < NN) { wBb[tid] = Bb[tid]; wCb[tid] = Cb[tid]; }
  if (tid < DD) { wdw[tid] = dw[tid]; }
  if (tid == 0) { wdb = dbp[0]; }

  const float* xb = x   + (size_t)b * DD * LL;
  float*       ob = out + (size_t)b * DD * LL;

  stage_tile(xb, xs[0], lbeg, tid);   // issue first tile (async if available)
  __syncthreads();

  // per-thread recurrence state (h=0 init: true start applied by K3)
  float h = 0.f, P = 1.f, a_dn = 0.f;
  int sd = 0, sn = 0;
  if (tid < DN) { sd = tid / NN; sn = tid % NN; a_dn = wA[tid]; }

  // loop-invariant WMMA weight fragments
  const int lane = tid & 31;
  const int fn   = lane & 15;
  const int kh   = (lane >> 4) * 2;
  const int mrow = ((tid >> 5) & 1) * 16;        // wave0 rows 0..15, wave1 16..31
  v2f fB[3], fC[3];
  #pragma unroll
  for (int c = 0; c < 3; ++c) {
    const int k0 = 4 * c + kh;
    float b0 = 0.f, b1 = 0.f, c0 = 0.f, c1 = 0.f;
    if (fn < NN) { b0 = wBw[fn][k0]; b1 = wBw[fn][k0 + 1];
                   c0 = wCw[fn][k0]; c1 = wCw[fn][k0 + 1]; }
    v2f tB; tB.x = b0; tB.y = b1; fB[c] = tB;
    v2f tC; tC.x = c0; tC.y = c1; fC[c] = tC;
  }
  const float biasB = (fn < NN) ? wBb[fn] : 0.f;
  const float biasC = (fn < NN) ? wCb[fn] : 0.f;

  int t = 0;
  for (int l0 = lbeg; l0 < lend; l0 += TL, t ^= 1) {
    wait_stage();
    __syncthreads();        // xs[t] visible; previous reduce complete

    if (tid < 64) {         // waves 0-1: Bm & Cm via f32 WMMA
      v8f aB = {}; v8f aC = {};
      #pragma unroll
      for (int c = 0; c < 3; ++c) {
        v2f av;
        const int m = mrow + (lane & 15);
        av.x = xs[t][m][4 * c + kh];
        av.y = xs[t][m][4 * c + kh + 1];
        aB = wmma4(av, fB[c], aB);
        aC = wmma4(av, fC[c], aC);
      }
      #pragma unroll
      for (int v = 0; v < 8; ++v) {
        const int m = mrow + v + ((lane >> 4) << 3);
        BmS[m][fn] = aB[v] + biasB;
        CmS[m][fn] = aC[v] + biasC;
      }
    } else if (tid >= 192 && tid < 192 + TL) {   // wave 6: softplus deltas
      const int i = tid - 192;
      float s = wdb;
      #pragma unroll
      for (int d = 0; d < DD; ++d) s = fmaf(xs[t][i][d], wdw[d], s);
      deltaS[i] = (s > 20.f) ? s : log1pf(__expf(s));
    }
    __syncthreads();

    if (tid < DN) {         // scan: precompute, then tight FMA chain (+ product P)
      const int rem = lend - l0;
      float ab[TL], dbx[TL];
      #pragma unroll
      for (int i = 0; i < TL; ++i) {
        const float dt = deltaS[i];
        const float dA = dt * a_dn;
        const float e  = __expf(dA);
        const float r  = __builtin_amdgcn_rcpf(dA + 1e-5f);
        const float v  = (e - 1.f) * r * (dt * xs[t][i][sd]) * BmS[i][sn];
        const bool ok  = (i < rem);
        ab[i]  = ok ? e : 1.f;
        dbx[i] = ok ? v : 0.f;
      }
      #pragma unroll
      for (int i = 0; i < TL; ++i) {
        h = fmaf(ab[i], h, dbx[i]);
        P *= ab[i];
        hist[i][tid] = h;
      }
    }
    __syncthreads();

    if (l0 + TL < lend) stage_tile(xb, xs[t ^ 1], l0 + TL, tid);  // overlap next tile

    for (int idx = tid; idx < DD * TL; idx += 256) {  // y_partial = sum_n h*Cm
      const int d = idx >> 5;
      const int i = idx & (TL - 1);
      const int l = l0 + i;
      if (l < lend) {
        const float* hr = &hist[i][d * NN];
        float y = 0.f;
        #pragma unroll
        for (int n = 0; n < NN; ++n) y = fmaf(hr[n], CmS[i][n], y);
        ob[(size_t)d * LL + l] = y;
      }
    }
  }

  if (tid < DN) {           // chunk summary for the carry scan
    const size_t o = ((size_t)b * NCH + cc) * DN + tid;
    hfin[o] = h;
    Pfin[o] = P;
  }
}

// ===============  K2: per-(b,d,n) carry scan over the 8 chunks  ===============
__global__ __launch_bounds__(160)
void ssm_carry_kernel(const float* __restrict__ h0,
                      const float* __restrict__ hfin,
                      const float* __restrict__ Pfin,
                      float* __restrict__ hstart)
{
  const int b = blockIdx.x;
  const int tid = threadIdx.x;
  if (tid >= DN) return;
  float h = h0[(size_t)b * DN + tid];
  hstart[((size_t)b * NCH) * DN + tid] = h;
  for (int c = 0; c < NCH - 1; ++c) {
    const size_t o = ((size_t)b * NCH + c) * DN + tid;
    h = fmaf(Pfin[o], h, hfin[o]);
    hstart[o + DN] = h;
  }
}

// ===========  K3: correction pass  y += sum_n (prod abar)*hstart*Cm  ==========
__global__ __launch_bounds__(256)
void ssm_fix_kernel(const float* __restrict__ x,
                    const float* __restrict__ A,
                    const float* __restrict__ Cw, const float* __restrict__ Cb,
                    const float* __restrict__ dw, const float* __restrict__ dbp,
                    const float* __restrict__ hstart,
                    float* __restrict__ out)
{
  __shared__ float xs[2][TL][DD + 1];
  __shared__ float CmS[TL][16];
  __shared__ float deltaS[TL];
  __shared__ float hist[TL][DN];
  __shared__ float wCw[NN][DD];
  __shared__ float wA[DN];
  __shared__ float wCb[NN], wdw[DD];
  __shared__ float wdb;

  const int cc  = blockIdx.x;
  const int b   = blockIdx.y;
  const int tid = threadIdx.x;
  const int lbeg = cc * CHL;
  const int lend = (lbeg + CHL < LL) ? lbeg + CHL : LL;

  if (tid < DN) {
    wCw[tid / DD][tid % DD] = Cw[tid];
    wA[tid] = A[tid];
  }
  if (tid < NN) { wCb[tid] = Cb[tid]; }
  if (tid < DD) { wdw[tid] = dw[tid]; }
  if (tid == 0) { wdb = dbp[0]; }

  const float* xb = x   + (size_t)b * DD * LL;
  float*       ob = out + (size_t)b * DD * LL;

  stage_tile(xb, xs[0], lbeg, tid);
  __syncthreads();

  float g = 0.f, a_dn = 0.f;
  int sd = 0;
  if (tid < DN) {
    sd   = tid / NN;
    g    = hstart[((size_t)b * NCH + cc) * DN + tid];  // true chunk-start state
    a_dn = wA[tid];
  }
  (void)sd;

  const int lane = tid & 31;
  const int fn   = lane & 15;
  const int kh   = (lane >> 4) * 2;
  const int mrow = ((tid >> 5) & 1) * 16;
  v2f fC[3];
  #pragma unroll
  for (int c = 0; c < 3; ++c) {
    const int k0 = 4 * c + kh;
    float c0 = 0.f, c1 = 0.f;
    if (fn < NN) { c0 = wCw[fn][k0]; c1 = wCw[fn][k0 + 1]; }
    v2f tC; tC.x = c0; tC.y = c1; fC[c] = tC;
  }
  const float biasC = (fn < NN) ? wCb[fn] : 0.f;

  int t = 0;
  for (int l0 = lbeg; l0 < lend; l0 += TL, t ^= 1) {
    wait_stage();
    __syncthreads();

    if (tid < 64) {         // waves 0-1: Cm via f32 WMMA
      v8f aC = {};
      #pragma unroll
      for (int c = 0; c < 3; ++c) {
        v2f av;
        const int m = mrow + (lane & 15);
        av.x = xs[t][m][4 * c + kh];
        av.y = xs[t][m][4 * c + kh + 1];
        aC = wmma4(av, fC[c], aC);
      }
      #pragma unroll
      for (int v = 0; v < 8; ++v) {
        const int m = mrow + v + ((lane >> 4) << 3);
        CmS[m][fn] = aC[v] + biasC;
      }
    } else if (tid >= 192 && tid < 192 + TL) {
      const int i = tid - 192;
      float s = wdb;
      #pragma unroll
      for (int d = 0; d < DD; ++d) s = fmaf(xs[t][i][d], wdw[d], s);
      deltaS[i] = (s > 20.f) ? s : log1pf(__expf(s));
    }
    __syncthreads();

    if (tid < DN) {         // multiplicative scan: g_l = hstart * prod abar
      const int rem = lend - l0;
      float ab[TL];
      #pragma unroll
      for (int i = 0; i < TL; ++i) {
        const float e = __expf(deltaS[i] * a_dn);
        ab[i] = (i < rem) ? e : 1.f;
      }
      #pragma unroll
      for (int i = 0; i < TL; ++i) {
        g *= ab[i];
        hist[i][tid] = g;
      }
    }
    __syncthreads();

    if (l0 + TL < lend) stage_tile(xb, xs[t ^ 1], l0 + TL, tid);

    for (int idx = tid; idx < DD * TL; idx += 256) {
      const int d = idx >> 5;
      const int i = idx & (TL - 1);
      const int l = l0 + i;
      if (l < lend) {
        const float* hr = &hist[i][d * NN];
        float y = 0.f;
        #pragma unroll
        for (int n = 0; n < NN; ++n) y = fmaf(hr[n], CmS[i][n], y);
        ob[(size_t)d * LL + l] += y;     // add correction to K1's partial
      }
    }
  }
}

extern "C" void kernel_launch(void* const* d_in, const int* in_sizes, int n_in,
                              void* d_out, int out_size, void* d_ws, size_t ws_size,
                              hipStream_t stream) {
  (void)in_sizes; (void)n_in; (void)out_size; (void)ws_size;
  const float* x  = (const float*)d_in[0];
  const float* h0 = (const float*)d_in[1];
  const float* A  = (const float*)d_in[2];
  const float* Bw = (const float*)d_in[3];
  const float* Bb = (const float*)d_in[4];
  const float* Cw = (const float*)d_in[5];
  const float* Cb = (const float*)d_in[6];
  const float* dw = (const float*)d_in[7];
  const float* db = (const float*)d_in[8];
  float* out = (float*)d_out;

  // workspace: chunk summaries (3 * 128*8*144 floats = 1.77 MB)
  float* hfin   = (float*)d_ws;
  float* Pfin   = hfin + (size_t)BSZ * NCH * DN;
  float* hstart = Pfin + (size_t)BSZ * NCH * DN;

  ssm_local_kernel<<<dim3(NCH, BSZ), dim3(256), 0, stream>>>(
      x, A, Bw, Bb, Cw, Cb, dw, db, out, hfin, Pfin);
  ssm_carry_kernel<<<dim3(BSZ), dim3(160), 0, stream>>>(h0, hfin, Pfin, hstart);
  ssm_fix_kernel<<<dim3(NCH, BSZ), dim3(256), 0, stream>>>(
      x, A, Cw, Cb, dw, db, hstart, out);
}